// NGCF_33792802685381
// MI455X (gfx1250) — compile-verified
//
#include <hip/hip_runtime.h>

#define NUM_USERS 100000
#define NUM_ITEMS 50000
#define NTOT      (NUM_USERS + NUM_ITEMS)   // 150000
#define EMB       64
#define NLAYERS   3
#define NEDGES    2400000
#define OUTW      (EMB * (NLAYERS + 1))     // 256
#define NEG_SLOPE 0.01f
#define NORM_EPS  1e-12f

typedef __attribute__((ext_vector_type(2))) float v2f;
typedef __attribute__((ext_vector_type(8))) float v8f;

// ---------------------------------------------------------------------------
// K0: build ego = concat(user_emb, item_emb); also write layer-0 slice of out.
// ---------------------------------------------------------------------------
__global__ void ngcf_init_kernel(const float* __restrict__ ue,
                                 const float* __restrict__ ie,
                                 float* __restrict__ ego,
                                 float* __restrict__ out) {
    long gid = (long)blockIdx.x * 256 + threadIdx.x;   // one float4 chunk
    if (gid >= (long)NTOT * 16) return;
    int row = (int)(gid >> 4);
    int c4  = (int)(gid & 15) * 4;
    float4 v;
    if (row < NUM_USERS)
        v = *(const float4*)(ue + (long)row * EMB + c4);
    else
        v = *(const float4*)(ie + (long)(row - NUM_USERS) * EMB + c4);
    *(float4*)(ego + (long)row * EMB  + c4) = v;
    *(float4*)(out + (long)row * OUTW + c4) = v;
}

// ---------------------------------------------------------------------------
// K1: zero side buffer (float4 fill)
// ---------------------------------------------------------------------------
__global__ void ngcf_zero_kernel(float4* __restrict__ p, long n4) {
    long i = (long)blockIdx.x * 256 + threadIdx.x;
    if (i < n4) p[i] = make_float4(0.f, 0.f, 0.f, 0.f);
}

// ---------------------------------------------------------------------------
// K2: COO SpMM scatter: side[row] += val * ego[col].  16 lanes per edge,
// float4 per lane -> global_atomic_add_f32 (L2-resident: ego+side < 192MB L2)
// ---------------------------------------------------------------------------
__global__ void ngcf_spmm_kernel(const int* __restrict__ rows,
                                 const int* __restrict__ cols,
                                 const float* __restrict__ vals,
                                 const float* __restrict__ ego,
                                 float* __restrict__ side) {
    long gid = (long)blockIdx.x * 256 + threadIdx.x;
    int e = (int)(gid >> 4);
    if (e >= NEDGES) return;
    int c4 = (int)(gid & 15) * 4;
    int r = rows[e];
    int c = cols[e];
    float v = vals[e];
    float4 s = *(const float4*)(ego + (long)c * EMB + c4);
    float* dst = side + (long)r * EMB + c4;
    atomicAdd(dst + 0, v * s.x);
    atomicAdd(dst + 1, v * s.y);
    atomicAdd(dst + 2, v * s.z);
    atomicAdd(dst + 3, v * s.w);
}

// ---------------------------------------------------------------------------
// K3: fused dual GEMM via V_WMMA_F32_16X16X4_F32 (exact fp32 math):
//   ego_new = leaky(side @ Wgc + bgc) + leaky((ego .* side) @ Wbi + bbi)
// One wave per 16-row tile. Weights staged in LDS. In-place ego update is
// safe: each wave reads only its own 16 rows before writing them.
//
// WMMA f32 16x16x4 operand mapping (ISA 7.12.2):
//   A (16x4):  lane l: a.x = A[l%16][2*(l/16)],   a.y = A[l%16][2*(l/16)+1]
//   B (4x16):  lane l: b.x = B[2*(l/16)][l%16],   b.y = B[2*(l/16)+1][l%16]
//   C/D:       vgpr r, lane l: D[r + 8*(l/16)][l%16]
// ---------------------------------------------------------------------------
__global__ __launch_bounds__(256) void ngcf_gemm_kernel(
        const float* __restrict__ side,
        float* __restrict__ ego,             // in: ego, out: ego_new (in place)
        const float* __restrict__ gw, const float* __restrict__ gb,
        const float* __restrict__ bw, const float* __restrict__ bb) {
    __shared__ float lgw[EMB * EMB];
    __shared__ float lbw[EMB * EMB];
    __shared__ float lgb[EMB];
    __shared__ float lbb[EMB];
    for (int i = threadIdx.x; i < EMB * EMB; i += 256) {
        lgw[i] = gw[i];
        lbw[i] = bw[i];
    }
    if (threadIdx.x < EMB) {
        lgb[threadIdx.x] = gb[threadIdx.x];
        lbb[threadIdx.x] = bb[threadIdx.x];
    }
    __syncthreads();

    const int wave = threadIdx.x >> 5;
    const int lane = threadIdx.x & 31;
    const int tile = blockIdx.x * 8 + wave;
    if (tile * 16 >= NTOT) return;           // uniform per wave: EXEC stays full

    const int half = lane >> 4;              // 0: lanes 0-15, 1: lanes 16-31
    const int lrow = lane & 15;
    const long rowbase = (long)(tile * 16 + lrow) * EMB;

    v8f ag[4] = {v8f{}, v8f{}, v8f{}, v8f{}};  // side @ Wgc accumulators
    v8f ab[4] = {v8f{}, v8f{}, v8f{}, v8f{}};  // (ego.*side) @ Wbi accumulators

    #pragma unroll
    for (int ks = 0; ks < 16; ++ks) {        // K = 64 in steps of 4
        const int k = ks * 4 + half * 2;
        v2f s2 = *(const v2f*)(side + rowbase + k);
        v2f e2 = *(const v2f*)(ego  + rowbase + k);
        v2f a1 = s2;
        v2f a2 = s2 * e2;
        #pragma unroll
        for (int t = 0; t < 4; ++t) {        // 4 output-column tiles of 16
            const int n = t * 16 + lrow;
            v2f b1, b2;
            b1.x = lgw[k * EMB + n];
            b1.y = lgw[(k + 1) * EMB + n];
            b2.x = lbw[k * EMB + n];
            b2.y = lbw[(k + 1) * EMB + n];
            ag[t] = __builtin_amdgcn_wmma_f32_16x16x4_f32(
                        false, a1, false, b1, (short)0, ag[t], false, false);
            ab[t] = __builtin_amdgcn_wmma_f32_16x16x4_f32(
                        false, a2, false, b2, (short)0, ab[t], false, false);
        }
    }

    // Epilogue: bias + leaky_relu on both branches, sum, store ego_new.
    #pragma unroll
    for (int t = 0; t < 4; ++t) {
        const int n = t * 16 + lrow;         // output column for this lane
        const float bgc = lgb[n];
        const float bbi = lbb[n];
        #pragma unroll
        for (int r = 0; r < 8; ++r) {
            float x1 = ag[t][r] + bgc;
            x1 = x1 > 0.f ? x1 : NEG_SLOPE * x1;
            float x2 = ab[t][r] + bbi;
            x2 = x2 > 0.f ? x2 : NEG_SLOPE * x2;
            const int row = tile * 16 + r + half * 8;
            ego[(long)row * EMB + n] = x1 + x2;
        }
    }
}

// ---------------------------------------------------------------------------
// K4: row L2-normalize ego_new into out[:, coloff:coloff+64].
// One wave per row, 2 floats/lane, shfl_xor reduction (wave32).
// ---------------------------------------------------------------------------
__global__ void ngcf_norm_kernel(const float* __restrict__ ego,
                                 float* __restrict__ out, int coloff) {
    const int wave = threadIdx.x >> 5;
    const int lane = threadIdx.x & 31;
    const int row = blockIdx.x * 8 + wave;
    if (row >= NTOT) return;
    v2f v = *(const v2f*)(ego + (long)row * EMB + lane * 2);
    float ss = v.x * v.x + v.y * v.y;
    #pragma unroll
    for (int m = 16; m >= 1; m >>= 1) ss += __shfl_xor(ss, m, 32);
    const float scale = 1.0f / fmaxf(sqrtf(ss), NORM_EPS);
    v2f o;
    o.x = v.x * scale;
    o.y = v.y * scale;
    *(v2f*)(out + (long)row * OUTW + coloff + lane * 2) = o;
}

// ---------------------------------------------------------------------------
extern "C" void kernel_launch(void* const* d_in, const int* in_sizes, int n_in,
                              void* d_out, int out_size, void* d_ws, size_t ws_size,
                              hipStream_t stream) {
    const int*   adj_rows = (const int*)d_in[0];
    const int*   adj_cols = (const int*)d_in[1];
    const float* adj_vals = (const float*)d_in[2];
    const float* user_emb = (const float*)d_in[3];
    const float* item_emb = (const float*)d_in[4];
    const float* gc_w     = (const float*)d_in[5];
    const float* gc_b     = (const float*)d_in[6];
    const float* bi_w     = (const float*)d_in[7];
    const float* bi_b     = (const float*)d_in[8];
    float* out = (float*)d_out;

    float* ego  = (float*)d_ws;                 // [NTOT, 64]  38.4 MB
    float* side = ego + (size_t)NTOT * EMB;     // [NTOT, 64]  38.4 MB

    const long init_threads = (long)NTOT * 16;
    ngcf_init_kernel<<<(int)((init_threads + 255) / 256), 256, 0, stream>>>(
        user_emb, item_emb, ego, out);

    const long n4 = (long)NTOT * 16;            // float4 chunks in side
    const long spmm_threads = (long)NEDGES * 16;
    const int gemm_blocks = (NTOT / 16 + 7) / 8;
    const int norm_blocks = (NTOT + 7) / 8;

    for (int l = 0; l < NLAYERS; ++l) {
        ngcf_zero_kernel<<<(int)((n4 + 255) / 256), 256, 0, stream>>>(
            (float4*)side, n4);
        ngcf_spmm_kernel<<<(int)((spmm_threads + 255) / 256), 256, 0, stream>>>(
            adj_rows, adj_cols, adj_vals, ego, side);
        ngcf_gemm_kernel<<<gemm_blocks, 256, 0, stream>>>(
            side, ego,
            gc_w + (size_t)l * EMB * EMB, gc_b + (size_t)l * EMB,
            bi_w + (size_t)l * EMB * EMB, bi_b + (size_t)l * EMB);
        ngcf_norm_kernel<<<norm_blocks, 256, 0, stream>>>(
            ego, out, EMB * (l + 1));
    }
}